// sGuidedFilter_35931696398900
// MI455X (gfx1250) — compile-verified
//
#include <hip/hip_runtime.h>
#include <math.h>

// Side-window guided filter for MI455X (gfx1250, wave32).
// All 9 conv kernels are rank-1 separable 0/1 masks -> implemented as banded
// f32 matmuls on the matrix pipe: V_WMMA_F32_16X16X4_F32 chains (K=32 total).
// One wave per 16x16 output tile; clamped 32x32 halo staged through LDS.
// Interior tiles use the Tensor Data Mover (tensor_load_to_lds) for staging;
// border tiles (replication clamp) use per-lane loads.
// Band-matrix WMMA fragments are generated as single v_cndmask_b32 ops from
// compile-time 32-bit lane masks (one instruction per fragment component).

#define IMG_H 2048
#define IMG_W 2048
#define RAD   8

typedef __attribute__((ext_vector_type(2))) float v2f;
typedef __attribute__((ext_vector_type(8))) float v8f;

typedef unsigned int u32x4 __attribute__((ext_vector_type(4)));
typedef int          i32x8 __attribute__((ext_vector_type(8)));
typedef int          i32x4 __attribute__((ext_vector_type(4)));

#if __has_builtin(__builtin_amdgcn_tensor_load_to_lds) && \
    __has_builtin(__builtin_amdgcn_s_wait_tensorcnt)
#define HAVE_TDM 1
#else
#define HAVE_TDM 0
#endif

__device__ __forceinline__ int clampi(int v, int lo, int hi) {
    return v < lo ? lo : (v > hi ? hi : v);
}

// Compile-time lane mask for one fragment component of the 0/1 band matrix.
// Bit l = value this lane contributes for K-chunk kb, component comp (0/1).
// wt 0: full window  -> u in [n, n+16]
// wt 1: left/top     -> u in [n, n+8]
// wt 2: right/bottom -> u in [n+8, n+16]
constexpr unsigned wmask(int wt, int kb, int comp) {
    unsigned m = 0;
    for (int l = 0; l < 32; ++l) {
        int u = kb * 4 + ((l >> 4) << 1) + comp;   // K index held by lane l
        int n = l & 15;                            // output row/col of lane l
        int lo = (wt == 2) ? n + 8 : n;
        int hi = (wt == 1) ? n + 8 : n + 16;
        if (u >= lo && u <= hi) m |= (1u << l);
    }
    return m;
}

// One-VALU fragment materialization: per-lane select of 1.0/0.0 from an SGPR
// lane mask (wave32 cndmask selector).
__device__ __forceinline__ float mask_to_float(unsigned m) {
    float f;
    asm("v_cndmask_b32 %0, 0, 1.0, %1" : "=v"(f) : "s"(m));
    return f;
}

template <int WT>
__device__ __forceinline__ v2f wfrag(int kb) {
    v2f w;
    w.x = mask_to_float(wmask(WT, kb, 0));
    w.y = mask_to_float(wmask(WT, kb, 1));
    return w;
}

// Clamped (replication-pad) per-lane load of the 32x32 halo tile.
__device__ __forceinline__ void load_tile32(const float* __restrict__ src,
                                            int gy0, int gx0, float* s, int lane) {
#pragma unroll
    for (int t = 0; t < 32; ++t) {
        int gy = clampi(gy0 - RAD + t, 0, IMG_H - 1);
        int gx = clampi(gx0 - RAD + lane, 0, IMG_W - 1);
        s[t * 32 + lane] = src[(size_t)gy * IMG_W + gx];
    }
}

#if HAVE_TDM
// TDM 2D tile load: 32x32 f32 tile at (gy0-8, gx0-8), row stride IMG_W.
// D# per CDNA5 ISA 8.3/8.4: group0 = {count, lds_addr, global_addr, type=2},
// group1 = {data_size=4B, tensor dims, tile dims 32x32, dim0 stride}.
__device__ __forceinline__ void tdm_load_tile32(const float* __restrict__ src,
                                                int gy0, int gx0, float* lds) {
    unsigned long long gaddr = (unsigned long long)(const void*)
        (src + (size_t)(gy0 - RAD) * IMG_W + (gx0 - RAD));
    unsigned int laddr = (unsigned int)(unsigned long long)(void*)lds;
    u32x4 g0;
    g0[0] = 1u;                                            // count=1 user D#
    g0[1] = laddr;                                         // lds_addr
    g0[2] = (unsigned int)(gaddr & 0xffffffffu);           // global_addr lo
    g0[3] = (unsigned int)((gaddr >> 32) & 0x01ffffffu)    // global_addr hi
            | (2u << 30);                                  // type=2 (image)
    i32x8 g1;
    g1[0] = 2 << 16;                                       // data_size=2 (4B)
    g1[1] = (IMG_W & 0xffff) << 16;                        // tensor_dim0 lo16
    g1[2] = (IMG_H & 0xffff) << 16;                        // dim0 hi=0 | dim1 lo16
    g1[3] = 32 << 16;                                      // dim1 hi=0 | tile_dim0=32
    g1[4] = 32;                                            // tile_dim1=32, tile_dim2=0
    g1[5] = IMG_W;                                         // tensor_dim0_stride lo32
    g1[6] = 0;                                             // stride hi | dim1_stride lo
    g1[7] = 0;
    i32x4 z4 = {};
    i32x8 z8 = {};
    __builtin_amdgcn_tensor_load_to_lds(g0, g1, z4, z4, z8, 0);
}
#endif

// Stage one 32x32 plane: TDM for interior tiles, clamped loads at borders.
__device__ __forceinline__ void stage_tile(const float* __restrict__ src,
                                           int gy0, int gx0, float* s,
                                           int lane, bool interior) {
#if HAVE_TDM
    if (interior) {
        tdm_load_tile32(src, gy0, gx0, s);
        __builtin_amdgcn_s_wait_tensorcnt(0);
        return;
    }
#endif
    (void)interior;
    load_tile32(src, gy0, gx0, s, lane);
}

// Horizontal pass: H(16 rows of block rb x 16 cols) = In(16x32) x W(32x16).
// Optional s1 multiplies elementwise (for x*x, x*y planes). Result -> outH.
template <int WT>
__device__ __forceinline__ void hpass(const float* s0, const float* s1, int rb,
                                      float* outH, int lane) {
    v8f acc = {};
    const int n = lane & 15;
    const int khalf = (lane >> 4) << 1;
    const int r = rb * 16 + n;
#pragma unroll
    for (int kb = 0; kb < 8; ++kb) {
        int k0 = kb * 4 + khalf;
        float a0 = s0[r * 32 + k0];
        float a1 = s0[r * 32 + k0 + 1];
        if (s1) { a0 *= s1[r * 32 + k0]; a1 *= s1[r * 32 + k0 + 1]; }
        v2f a;
        a.x = a0; a.y = a1;
        acc = __builtin_amdgcn_wmma_f32_16x16x4_f32(
            false, a, false, wfrag<WT>(kb), (short)0, acc, false, false);
    }
#pragma unroll
    for (int k = 0; k < 8; ++k) {
        int row = rb * 16 + k + ((lane >> 4) << 3);   // D layout: M=k (+8 hi half)
        outH[row * 16 + n] = acc[k];
    }
}

// Vertical pass: Out(16x16) = S(16x32) x H(32x16).
template <int VT>
__device__ __forceinline__ v8f vpass(const float* sHp, int lane) {
    v8f acc = {};
    const int n = lane & 15;
    const int khalf = (lane >> 4) << 1;
#pragma unroll
    for (int kb = 0; kb < 8; ++kb) {
        int k0 = kb * 4 + khalf;
        v2f b;
        b.x = sHp[k0 * 16 + n];              // H[u][n]
        b.y = sHp[(k0 + 1) * 16 + n];
        acc = __builtin_amdgcn_wmma_f32_16x16x4_f32(
            false, wfrag<VT>(kb), false, b, (short)0, acc, false, false);
    }
    return acc;
}

// ---------------- Kernel 1: guided-filter coefficients A, b ----------------
__global__ __launch_bounds__(32)
void swf_coeff_kernel(const float* __restrict__ x, const float* __restrict__ y,
                      float* __restrict__ Aout, float* __restrict__ Bout) {
    __shared__ float sX[32 * 32];
    __shared__ float sY[32 * 32];
    __shared__ float sH[32 * 16];

    const int lane = threadIdx.x;
    const int tilesX = IMG_W / 16;
    const int tilesPerCh = (IMG_H / 16) * tilesX;
    int bid = blockIdx.x;
    int c = bid / tilesPerCh;
    int t = bid - c * tilesPerCh;
    int ty = t / tilesX, tx = t - ty * tilesX;
    int gy0 = ty * 16, gx0 = tx * 16;
    bool interior = (gy0 >= RAD) && (gx0 >= RAD) &&
                    (gy0 + 16 + RAD <= IMG_H) && (gx0 + 16 + RAD <= IMG_W);

    const float* xc = x + (size_t)c * IMG_H * IMG_W;
    const float* yc = y + (size_t)c * IMG_H * IMG_W;

    stage_tile(xc, gy0, gx0, sX, lane, interior);
    stage_tile(yc, gy0, gx0, sY, lane, interior);
    __syncthreads();

    v8f Sx, Sy, Sxx, Sxy;
    hpass<0>(sX, nullptr, 0, sH, lane); hpass<0>(sX, nullptr, 1, sH, lane);
    __syncthreads(); Sx = vpass<0>(sH, lane); __syncthreads();
    hpass<0>(sY, nullptr, 0, sH, lane); hpass<0>(sY, nullptr, 1, sH, lane);
    __syncthreads(); Sy = vpass<0>(sH, lane); __syncthreads();
    hpass<0>(sX, sX, 0, sH, lane); hpass<0>(sX, sX, 1, sH, lane);
    __syncthreads(); Sxx = vpass<0>(sH, lane); __syncthreads();
    hpass<0>(sX, sY, 0, sH, lane); hpass<0>(sX, sY, 1, sH, lane);
    __syncthreads(); Sxy = vpass<0>(sH, lane);

    const float inv = 1.0f / 289.0f;
#pragma unroll
    for (int k = 0; k < 8; ++k) {
        float mx = Sx[k] * inv, my = Sy[k] * inv;
        float varx = Sxx[k] * inv - mx * mx;
        float cov  = Sxy[k] * inv - mx * my;
        float Ae = cov / (varx + 1e-6f);
        float Be = my - Ae * mx;
        int gy = gy0 + k + ((lane >> 4) << 3);
        int gx = gx0 + (lane & 15);
        size_t idx = ((size_t)c * IMG_H + gy) * IMG_W + gx;
        Aout[idx] = Ae;
        Bout[idx] = Be;
    }
}

// ------------- Kernel 2: 8 side-window convs, argmin, finalize -------------
__global__ __launch_bounds__(32)
void swf_apply_kernel(const float* __restrict__ x, const float* __restrict__ Ain,
                      const float* __restrict__ Bin, float* __restrict__ out) {
    __shared__ float sP [32 * 32];
    __shared__ float sHf[32 * 16];
    __shared__ float sHl[32 * 16];
    __shared__ float sHr[32 * 16];

    const int lane = threadIdx.x;
    const int tilesX = IMG_W / 16;
    const int tilesPerCh = (IMG_H / 16) * tilesX;
    int bid = blockIdx.x;
    int c = bid / tilesPerCh;
    int t = bid - c * tilesPerCh;
    int ty = t / tilesX, tx = t - ty * tilesX;
    int gy0 = ty * 16, gx0 = tx * 16;
    bool interior = (gy0 >= RAD) && (gx0 >= RAD) &&
                    (gy0 + 16 + RAD <= IMG_H) && (gx0 + 16 + RAD <= IMG_W);

    const float* xc = x   + (size_t)c * IMG_H * IMG_W;
    const float* Ac = Ain + (size_t)c * IMG_H * IMG_W;
    const float* Bc = Bin + (size_t)c * IMG_H * IMG_W;

    const float ih = 1.0f / 153.0f;   // (r+1)*k
    const float iq = 1.0f / 81.0f;    // (r+1)^2
    const int n = lane & 15;
    const int hi8 = (lane >> 4) << 3;

    // ---- plane A ----
    stage_tile(Ac, gy0, gx0, sP, lane, interior);
    __syncthreads();
    hpass<0>(sP, nullptr, 0, sHf, lane); hpass<0>(sP, nullptr, 1, sHf, lane);
    hpass<1>(sP, nullptr, 0, sHl, lane); hpass<1>(sP, nullptr, 1, sHl, lane);
    hpass<2>(sP, nullptr, 0, sHr, lane); hpass<2>(sP, nullptr, 1, sHr, lane);
    __syncthreads();
    v8f Tf = vpass<1>(sHf, lane), Bf = vpass<2>(sHf, lane);
    v8f Tl = vpass<1>(sHl, lane), Bl = vpass<2>(sHl, lane);
    v8f Tr = vpass<1>(sHr, lane), Br = vpass<2>(sHr, lane);

    v8f dA[8];
#pragma unroll
    for (int k = 0; k < 8; ++k) {
        float hlC = sHl[(k + hi8 + 8) * 16 + n];   // center row of H (vr = i+8)
        float hrC = sHr[(k + hi8 + 8) * 16 + n];
        dA[0][k] = (Tl[k] + Bl[k] - hlC) * ih;  // L
        dA[1][k] = (Tr[k] + Br[k] - hrC) * ih;  // R
        dA[2][k] = Tf[k] * ih;                  // U
        dA[3][k] = Bf[k] * ih;                  // D
        dA[4][k] = Tl[k] * iq;                  // NW
        dA[5][k] = Tr[k] * iq;                  // NE
        dA[6][k] = Bl[k] * iq;                  // SW
        dA[7][k] = Br[k] * iq;                  // SE
    }
    __syncthreads();

    // ---- plane b ----
    stage_tile(Bc, gy0, gx0, sP, lane, interior);
    __syncthreads();
    hpass<0>(sP, nullptr, 0, sHf, lane); hpass<0>(sP, nullptr, 1, sHf, lane);
    hpass<1>(sP, nullptr, 0, sHl, lane); hpass<1>(sP, nullptr, 1, sHl, lane);
    hpass<2>(sP, nullptr, 0, sHr, lane); hpass<2>(sP, nullptr, 1, sHr, lane);
    __syncthreads();
    Tf = vpass<1>(sHf, lane); Bf = vpass<2>(sHf, lane);
    Tl = vpass<1>(sHl, lane); Bl = vpass<2>(sHl, lane);
    Tr = vpass<1>(sHr, lane); Br = vpass<2>(sHr, lane);

#pragma unroll
    for (int k = 0; k < 8; ++k) {
        int gy = gy0 + k + hi8;
        int gx = gx0 + n;
        size_t idx = ((size_t)c * IMG_H + gy) * IMG_W + gx;
        float xv = xc[(size_t)gy * IMG_W + gx];

        float hlC = sHl[(k + hi8 + 8) * 16 + n];
        float hrC = sHr[(k + hi8 + 8) * 16 + n];
        float dB[8];
        dB[0] = (Tl[k] + Bl[k] - hlC) * ih;
        dB[1] = (Tr[k] + Br[k] - hrC) * ih;
        dB[2] = Tf[k] * ih;
        dB[3] = Bf[k] * ih;
        dB[4] = Tl[k] * iq;
        dB[5] = Tr[k] * iq;
        dB[6] = Bl[k] * iq;
        dB[7] = Br[k] * iq;

        // d_dir = dirA*x + dirB - x ; first-min argmin over |d| (matches argmin)
        float best = dA[0][k] * xv + dB[0] - xv;
#pragma unroll
        for (int dir = 1; dir < 8; ++dir) {
            float d = dA[dir][k] * xv + dB[dir] - xv;
            if (fabsf(d) < fabsf(best)) best = d;
        }
        float res = truncf(best + xv);
        res = fminf(fmaxf(res, 0.0f), 255.0f);
        out[idx] = res;
    }
}

extern "C" void kernel_launch(void* const* d_in, const int* in_sizes, int n_in,
                              void* d_out, int out_size, void* d_ws, size_t ws_size,
                              hipStream_t stream) {
    (void)in_sizes; (void)n_in; (void)out_size; (void)ws_size;
    const float* x = (const float*)d_in[0];
    const float* y = (const float*)d_in[1];
    float* Aplane = (float*)d_ws;                                  // 3*H*W f32
    float* Bplane = Aplane + (size_t)3 * IMG_H * IMG_W;            // 3*H*W f32
    float* out = (float*)d_out;

    const int blocks = 3 * (IMG_H / 16) * (IMG_W / 16);            // 49152
    swf_coeff_kernel<<<blocks, 32, 0, stream>>>(x, y, Aplane, Bplane);
    swf_apply_kernel<<<blocks, 32, 0, stream>>>(x, Aplane, Bplane, out);
}